// LinearGaussianSCM_4672924418502
// MI455X (gfx1250) — compile-verified
//
#include <hip/hip_runtime.h>
#include <hip/hip_bf16.h>
#include <math.h>

typedef float v2f __attribute__((ext_vector_type(2)));
typedef float v8f __attribute__((ext_vector_type(8)));

#define D 128
#define NSRC 8
#define STRIDE 132              // LDS row stride (pads banks: 132 % 64 = 4; 528B = 33*16B, keeps 16B align)
#define EPS 1e-6f
#define LOG_2PI 1.8378770664093453f

// ------------------------------------------------ async global->LDS (CDNA5)
__device__ __forceinline__ void async_copy_b128(unsigned lds_byte_off,
                                                const void* gaddr) {
    asm volatile("global_load_async_to_lds_b128 %0, %1, off"
                 :: "v"(lds_byte_off), "v"(gaddr) : "memory");
}
__device__ __forceinline__ void wait_asynccnt0() {
    asm volatile("s_wait_asynccnt 0" ::: "memory");
}

// ---------------------------------------------------------------- zero output
__global__ void zero_out_kernel(float* out) {
    if (threadIdx.x == 0) out[0] = 0.0f;
}

// ------------------------------------------------- T = (I - W)^-1, col-major
// I - W is unit lower triangular; each column is an independent forward
// substitution: T[c][c]=1, T[r][c] = sum_{k=c..r-1} W[r][k] * T[k][c].
__global__ void build_T_kernel(const float* __restrict__ W,
                               float* __restrict__ Tcm) {
    extern __shared__ float smem[];        // 128 cols * stride 129
    const int c = threadIdx.x;             // 128 threads, one column each
    float* myc = smem + c * 129;
    #pragma unroll 4
    for (int r = 0; r < D; ++r) myc[r] = 0.0f;
    myc[c] = 1.0f;
    for (int r = c + 1; r < D; ++r) {
        float s = 0.0f;
        for (int k = c; k < r; ++k) s += W[r * D + k] * myc[k];
        myc[r] = s;
    }
    for (int r = 0; r < D; ++r) Tcm[c * D + r] = myc[r];   // column-major out
}

// ------------------------- Sigma[s][j][k] = sum_i T[i][j] sig2[s][i] T[i][k]
__global__ void build_Sigma_kernel(const float* __restrict__ Tcm,
                                   const float* __restrict__ log_sigma2,
                                   float* __restrict__ Sigma) {
    __shared__ float sig[D];
    const int s   = blockIdx.x;
    const int tid = threadIdx.x;
    if (tid < D) sig[tid] = fmaxf(expf(log_sigma2[s * D + tid]), EPS);
    __syncthreads();
    for (int e = tid; e < D * D; e += blockDim.x) {
        const int j = e >> 7, k = e & 127;
        const float* tj = Tcm + j * D;     // column j of T, contiguous in i
        const float* tk = Tcm + k * D;
        float acc = 0.0f;
        #pragma unroll 4
        for (int i = 0; i < D; ++i) acc += tj[i] * sig[i] * tk[i];
        Sigma[s * D * D + e] = acc;
    }
}

// --------------------------------------------- per-sample masked Cholesky LL
__global__ __launch_bounds__(256)
void masked_chol_ll_kernel(const float* __restrict__ data,
                           const float* __restrict__ mu,
                           const float* __restrict__ wgt,
                           const int*   __restrict__ sources,
                           const unsigned char* __restrict__ miss,
                           const float* __restrict__ Sigma,
                           float* __restrict__ out) {
    extern __shared__ float smem[];
    float* A    = smem;                      // D x STRIDE
    float* xcL  = smem + D * STRIDE;         // D
    float* obsL = xcL + D;                   // D
    float* sh   = obsL + D;                  // sh[0] = 1/diag

    const int n      = blockIdx.x;
    const int tid    = threadIdx.x;
    const int lane   = tid & 31;
    const int waveId = __builtin_amdgcn_readfirstlane(tid >> 5); // scalar!

    const int src = sources[n];
    const float* Sg = Sigma + (size_t)src * D * D;
    const unsigned char* mm = miss + (size_t)n * D;

    if (tid < D) {
        const float o = mm[tid] ? 0.0f : 1.0f;
        obsL[tid] = o;
        xcL[tid]  = (data[(size_t)n * D + tid] - mu[tid]) * o;
    }

    // ---- async copy Sigma[src] (64KB, L2-resident) straight into LDS ------
    // 16B per lane; both LDS (row stride 528B) and global (512B) are 16B-aligned.
    {
        const unsigned dynbase = __builtin_amdgcn_groupstaticsize();
        #pragma unroll
        for (int it = 0; it < 16; ++it) {
            const int e  = tid + it * 256;     // 0..4095 chunks of 4 floats
            const int r  = e >> 5;
            const int c4 = (e & 31) * 4;
            const unsigned lds_off = dynbase +
                (unsigned)((const char*)(A + r * STRIDE + c4) - (const char*)smem);
            async_copy_b128(lds_off, (const void*)(Sg + r * D + c4));
        }
        wait_asynccnt0();
    }
    __syncthreads();

    // mask in place: Sig_t = obs_r*obs_c*(Sigma + eps*I) + diag(1-obs)
    for (int e = tid; e < D * D; e += 256) {
        const int r = e >> 7, c = e & 127;
        const float dg = (r == c) ? 1.0f : 0.0f;
        const float orr = obsL[r], oc = obsL[c];
        const float v = A[r * STRIDE + c];
        A[r * STRIDE + c] = orr * oc * (v + dg * EPS) + dg * (1.0f - orr);
    }

    // -------- blocked right-looking Cholesky, block = 16, 8 k-steps --------
    for (int kb = 0; kb < 8; ++kb) {
        const int c0 = kb * 16;
        // panel factorization of columns c0..c0+15 over all rows below
        for (int j = 0; j < 16; ++j) {
            const int c = c0 + j;
            __syncthreads();
            if (tid == 0) {
                float d = sqrtf(fmaxf(A[c * STRIDE + c], 1e-30f));
                A[c * STRIDE + c] = d;
                sh[0] = 1.0f / d;
            }
            __syncthreads();
            const float invd = sh[0];
            for (int r = c + 1 + tid; r < D; r += 256)
                A[r * STRIDE + c] *= invd;
            __syncthreads();
            // rank-1 down-date restricted to remaining panel columns
            for (int c2 = c + 1; c2 <= c0 + 15; ++c2) {
                const float lc2 = A[c2 * STRIDE + c];
                for (int r = c2 + tid; r < D; r += 256)
                    A[r * STRIDE + c2] -= A[r * STRIDE + c] * lc2;
            }
        }
        __syncthreads();
        // trailing update: A[ib][jb] -= L[ib][kb] * L[jb][kb]^T via f32 WMMA
        if (kb < 7) {
            const int m16 = lane & 15;
            const int h   = lane >> 4;       // selects k-pair / row-half
            int idx = 0;
            for (int jb = kb + 1; jb < 8; ++jb) {
                for (int ib = jb; ib < 8; ++ib, ++idx) {
                    if ((idx & 7) != waveId) continue;   // scalar branch
                    // C fragment: VGPR r holds M=r (lanes 0-15) / M=r+8 (16-31)
                    const int crow = ib * 16 + h * 8;
                    const int ccol = jb * 16 + m16;
                    v8f cf;
                    #pragma unroll
                    for (int r = 0; r < 8; ++r)
                        cf[r] = A[(crow + r) * STRIDE + ccol];
                    const int arow = ib * 16 + m16;
                    const int brow = jb * 16 + m16;
                    #pragma unroll
                    for (int t = 0; t < 4; ++t) {
                        const int k0 = c0 + t * 4 + h * 2;
                        v2f a, b;
                        a[0] = -A[arow * STRIDE + k0];     // negate: C -= L*L^T
                        a[1] = -A[arow * STRIDE + k0 + 1];
                        b[0] =  A[brow * STRIDE + k0];     // B = L[jb]^T layout
                        b[1] =  A[brow * STRIDE + k0 + 1];
                        cf = __builtin_amdgcn_wmma_f32_16x16x4_f32(
                                 false, a, false, b, (short)0, cf, false, false);
                    }
                    #pragma unroll
                    for (int r = 0; r < 8; ++r)
                        A[(crow + r) * STRIDE + ccol] = cf[r];
                }
            }
        }
        __syncthreads();
    }

    // ---------------- forward solve L z = xc, quad, logdet (wave 0) ---------
    if (waveId == 0) {
        float acc0 = 0.f, acc1 = 0.f, acc2 = 0.f, acc3 = 0.f, q = 0.f;
        for (int c = 0; c < D; ++c) {
            const int owner = c & 31;
            float zc = 0.0f;
            if (lane == owner) {
                const int w = c >> 5;
                const float a = (w == 0) ? acc0 : (w == 1) ? acc1
                              : (w == 2) ? acc2 : acc3;
                zc = (xcL[c] - a) / A[c * STRIDE + c];
                q += zc * zc;
            }
            zc = __shfl(zc, owner, 32);
            if (lane      > c) acc0 += A[ lane        * STRIDE + c] * zc;
            if (lane + 32 > c) acc1 += A[(lane + 32)  * STRIDE + c] * zc;
            if (lane + 64 > c) acc2 += A[(lane + 64)  * STRIDE + c] * zc;
            if (lane + 96 > c) acc3 += A[(lane + 96)  * STRIDE + c] * zc;
        }
        float ld = 0.0f, dob = 0.0f;
        for (int c = lane; c < D; c += 32) {
            ld  += logf(A[c * STRIDE + c]);
            dob += obsL[c];
        }
        #pragma unroll
        for (int off = 16; off > 0; off >>= 1) {
            q   += __shfl_xor(q,   off, 32);
            ld  += __shfl_xor(ld,  off, 32);
            dob += __shfl_xor(dob, off, 32);
        }
        if (lane == 0) {
            const float ll = -0.5f * (q + 2.0f * ld + dob * LOG_2PI);
            atomicAdd(out, ll * wgt[n]);
        }
    }
}

// ---------------------------------------------------------------------------
extern "C" void kernel_launch(void* const* d_in, const int* in_sizes, int n_in,
                              void* d_out, int out_size, void* d_ws, size_t ws_size,
                              hipStream_t stream) {
    const float* data   = (const float*)d_in[0];   // [N, D]
    const float* W      = (const float*)d_in[1];   // [D, D]
    const float* logs2  = (const float*)d_in[2];   // [S, D]
    const float* mu     = (const float*)d_in[3];   // [D]
    const float* wgt    = (const float*)d_in[4];   // [N]
    const int*   srcs   = (const int*)d_in[5];     // [N]
    const unsigned char* miss = (const unsigned char*)d_in[6]; // [N, D] bool
    float* out = (float*)d_out;

    float* Tcm   = (float*)d_ws;                   // D*D floats, column-major
    float* Sigma = Tcm + D * D;                    // S*D*D floats

    const int N = in_sizes[0] / D;                 // 4096

    zero_out_kernel<<<1, 32, 0, stream>>>(out);
    build_T_kernel<<<1, D, 128 * 129 * sizeof(float), stream>>>(W, Tcm);
    build_Sigma_kernel<<<NSRC, 256, 0, stream>>>(Tcm, logs2, Sigma);

    const size_t lds = (size_t)(D * STRIDE + D + D + 8) * sizeof(float); // ~68.6KB
    masked_chol_ll_kernel<<<N, 256, lds, stream>>>(data, mu, wgt, srcs, miss,
                                                   Sigma, out);
}